// GCN_60627758350407
// MI455X (gfx1250) — compile-verified
//
#include <hip/hip_runtime.h>
#include <hip/hip_bf16.h>
#include <math.h>

// ---------------------------------------------------------------------------
// GCN pipeline for MI455X (gfx1250, wave32, WMMA).
// eigh() eliminated: h = 1*c^T + diag(d-c)  =>
//   V h V^T = u u^T P2(L) + (P1(L) - P2(L)),
// P1/P2 degree-4 scalar polynomials -> matrix polynomials in L; only
// lambda_max (power iteration) + a rank-1 term remain.
// Big GEMMs: LDS-tiled, double-buffered V_WMMA_F32_16X16X4_F32 with 2x2
// register blocking per wave; LDS staging uses gfx1250 async global->LDS
// copies (GLOBAL_LOAD_ASYNC_TO_LDS_B128 + s_wait_asynccnt).
// ---------------------------------------------------------------------------

typedef float v2f __attribute__((ext_vector_type(2)));
typedef float v8f __attribute__((ext_vector_type(8)));
typedef int   v4i __attribute__((ext_vector_type(4)));

constexpr int NG   = 4096;      // N_GENES
constexpr int BAT  = 128;       // BATCH
constexpr int NC   = 10;        // NUM_CLASSES
constexpr int POOL = 8;
constexpr int NPOW = 32;        // power-iteration steps

// ---------------- gfx1250 async global->LDS copy (guarded) ----------------
#if __has_builtin(__builtin_amdgcn_global_load_async_to_lds_b128) && \
    __has_builtin(__builtin_amdgcn_s_wait_asynccnt)
#define HAVE_ASYNC_LDS 1
#else
#define HAVE_ASYNC_LDS 0
#endif

__device__ __forceinline__ void cp16_g2lds(const float* g, float* l) {
#if HAVE_ASYNC_LDS
  typedef __attribute__((address_space(1))) v4i* gp_t;   // global int4*
  typedef __attribute__((address_space(3))) v4i* lp_t;   // LDS int4*
  __builtin_amdgcn_global_load_async_to_lds_b128(
      (gp_t)(unsigned long long)(g),
      (lp_t)(unsigned int)(unsigned long long)(l), 0, 0);
#else
  *(float4*)l = *(const float4*)g;
#endif
}

__device__ __forceinline__ void wait_async_copies() {
#if HAVE_ASYNC_LDS
  __builtin_amdgcn_s_wait_asynccnt(0);
#endif
}

// ---------------- wave32 reductions ----------------
__device__ inline float waveSum(float v) {
#pragma unroll
  for (int off = 16; off > 0; off >>= 1) v += __shfl_xor(v, off, 32);
  return v;
}
__device__ inline float waveMax(float v) {
#pragma unroll
  for (int off = 16; off > 0; off >>= 1) v = fmaxf(v, __shfl_xor(v, off, 32));
  return v;
}

// ---------------- degree / Laplacian ----------------
__global__ void deg_kernel(const float* __restrict__ A, float* dinv, float* dsq) {
  int wave = (blockIdx.x * blockDim.x + threadIdx.x) >> 5;
  int lane = threadIdx.x & 31;
  if (wave >= NG) return;
  const float* row = A + (size_t)wave * NG;
  float cnt = 0.f;
  for (int j = lane; j < NG; j += 32) cnt += (row[j] != 0.0f) ? 1.f : 0.f;
  cnt = waveSum(cnt);
  if (lane == 0) { dinv[wave] = rsqrtf(cnt); dsq[wave] = sqrtf(cnt); }
}

__global__ void buildL_kernel(const float* __restrict__ A, const float* __restrict__ dinv,
                              const float* __restrict__ dsq, float* __restrict__ L) {
  size_t idx = (size_t)blockIdx.x * blockDim.x + threadIdx.x;
  if (idx >= (size_t)NG * NG) return;
  int i = (int)(idx >> 12);
  int j = (int)(idx & (NG - 1));
  float v = dinv[i] * A[idx] * dsq[j];
  if (i == j) v += 1.0f;
  L[idx] = v;
}

// ---------------- power iteration (lambda_max) ----------------
__global__ void fill_kernel(float* p, float val, int n) {
  int i = blockIdx.x * blockDim.x + threadIdx.x;
  if (i < n) p[i] = val;
}

__global__ void matvec_norm_kernel(const float* __restrict__ M, const float* __restrict__ v,
                                   float* __restrict__ y, float* norm2slot) {
  int wave = (blockIdx.x * blockDim.x + threadIdx.x) >> 5;
  int lane = threadIdx.x & 31;
  if (wave >= NG) return;
  const float* row = M + (size_t)wave * NG;
  float acc = 0.f;
  for (int j = lane; j < NG; j += 32) acc = fmaf(row[j], v[j], acc);
  acc = waveSum(acc);
  if (lane == 0) { y[wave] = acc; atomicAdd(norm2slot, acc * acc); }
}

__global__ void vec_scale_kernel(const float* __restrict__ y, float* __restrict__ v,
                                 const float* norm2slot) {
  int i = blockIdx.x * blockDim.x + threadIdx.x;
  if (i >= NG) return;
  float inv = rsqrtf(norm2slot[0] + 1e-30f);
  v[i] = y[i] * inv;
}

// ---------------- Chebyshev -> monomial-in-L coefficients ----------------
__global__ void coef_kernel(const float* __restrict__ beta, const float* norm2last,
                            float* __restrict__ cf) {
  if (threadIdx.x != 0 || blockIdx.x != 0) return;
  float lam = sqrtf(norm2last[0] + 1e-30f);
  float b0 = beta[0], b1 = beta[1], b2 = beta[2], b3 = beta[3], b4 = beta[4];
  float m0 = b0 - b2 + b4;
  float m1 = b1 - 3.f * b3;
  float m2 = 2.f * b2 - 8.f * b4;
  float m3 = 4.f * b3;
  float m4 = 8.f * b4;
  float q0 = m0 - m1 + m2 - m3 + m4;
  float q1 = m1 - 2.f * m2 + 3.f * m3 - 4.f * m4;
  float q2 = m2 - 3.f * m3 + 6.f * m4;
  float q3 = m3 - 4.f * m4;
  float q4 = m4;
  float al = 2.f / lam;
  float a0 = 1.f, a1 = al, a2 = al * al, a3 = a2 * al, a4 = a2 * a2;
  cf[0] = (q0 - m0) * a0;  cf[8]  = m0 * a0;
  cf[1] = (q1 - m1) * a1;  cf[9]  = m1 * a1;
  cf[2] = (q2 - m2) * a2;  cf[10] = m2 * a2;
  cf[3] = (q3 - m3) * a3;  cf[11] = m3 * a3;
  cf[4] = (q4 - m4) * a4;  cf[12] = m4 * a4;
  cf[13] = lam;
}

// ============ tiled WMMA f32 GEMM: 64x64 C-tile, LDS double-buffered =======
// Block: 128 threads = 4 waves; wave (wm,wn) owns a 32x32 register block
// (2x2 v8f accumulators). K staged in 16-wide panels.
// LDS pitches: A -> 20 floats (80B), B -> 80 floats (320B): 16B-aligned rows,
// conflict-free ds_read patterns for the WMMA fragment layouts.
#define TBM 64
#define TBN 64
#define TBK 16
#define TPA 20
#define TPB 80

__global__ __launch_bounds__(128)
void wmma_gemm_tiled(const float* __restrict__ A, const float* __restrict__ B,
                     float* __restrict__ D, const float* __restrict__ bias,
                     int M, int N, int K, int relu) {
  __shared__ float As[2][TBM][TPA];
  __shared__ float Bs[2][TBK][TPB];

  const int tid  = threadIdx.x;
  const int lane = tid & 31;
  const int wave = tid >> 5;
  const int wm   = wave >> 1;
  const int wn   = wave & 1;
  const int half = lane >> 4;
  const int l16  = lane & 15;

  const int mBlock = blockIdx.y * TBM;
  const int nBlock = blockIdx.x * TBN;

  v8f acc[2][2];
#pragma unroll
  for (int mi = 0; mi < 2; ++mi)
#pragma unroll
    for (int ni = 0; ni < 2; ++ni)
#pragma unroll
      for (int r = 0; r < 8; ++r) acc[mi][ni][r] = 0.f;

  const int KB = K / TBK;

  auto issue = [&](int buf, int kb) {
    // A panel: TBM rows x TBK cols = 256 16B chunks
#pragma unroll
    for (int c = tid; c < TBM * 4; c += 128) {
      int row = c >> 2, q = c & 3;
      cp16_g2lds(A + (size_t)(mBlock + row) * K + kb * TBK + q * 4,
                 &As[buf][row][q * 4]);
    }
    // B panel: TBK rows x TBN cols = 256 16B chunks
#pragma unroll
    for (int c = tid; c < TBK * (TBN / 4); c += 128) {
      int row = c >> 4, q = c & 15;
      cp16_g2lds(B + (size_t)(kb * TBK + row) * N + nBlock + q * 4,
                 &Bs[buf][row][q * 4]);
    }
  };

  auto compute = [&](int buf) {
#pragma unroll
    for (int ks = 0; ks < TBK; ks += 4) {
      int ka = ks + 2 * half;
      v2f af[2], bf[2];
#pragma unroll
      for (int mi = 0; mi < 2; ++mi) {
        const float* p = &As[buf][wm * 32 + mi * 16 + l16][ka];
        af[mi].x = p[0];
        af[mi].y = p[1];
      }
#pragma unroll
      for (int ni = 0; ni < 2; ++ni) {
        int col = wn * 32 + ni * 16 + l16;
        bf[ni].x = Bs[buf][ka][col];
        bf[ni].y = Bs[buf][ka + 1][col];
      }
#pragma unroll
      for (int mi = 0; mi < 2; ++mi)
#pragma unroll
        for (int ni = 0; ni < 2; ++ni)
          acc[mi][ni] = __builtin_amdgcn_wmma_f32_16x16x4_f32(
              false, af[mi], false, bf[ni], (short)0, acc[mi][ni], false, false);
    }
  };

  issue(0, 0);
  wait_async_copies();
  __syncthreads();
  for (int kb = 0; kb < KB; ++kb) {
    int cur = kb & 1;
    if (kb + 1 < KB) issue(cur ^ 1, kb + 1);
    compute(cur);
    wait_async_copies();
    __syncthreads();
  }

  float bv[2];
#pragma unroll
  for (int ni = 0; ni < 2; ++ni) {
    int col = nBlock + wn * 32 + ni * 16 + l16;
    bv[ni] = bias ? bias[col] : 0.f;
  }
#pragma unroll
  for (int mi = 0; mi < 2; ++mi)
#pragma unroll
    for (int ni = 0; ni < 2; ++ni) {
      int col = nBlock + wn * 32 + ni * 16 + l16;
#pragma unroll
      for (int r = 0; r < 8; ++r) {
        int row = mBlock + wm * 32 + mi * 16 + r + 8 * half;
        float v = acc[mi][ni][r] + bv[ni];
        if (relu) v = fmaxf(v, 0.f);
        D[(size_t)row * N + col] = v;
      }
    }
}

// ---------------- simple WMMA GEMM (small-N cases) ----------------
__global__ void wmma_gemm_f32(const float* __restrict__ A, const float* __restrict__ B,
                              float* __restrict__ D, const float* __restrict__ bias,
                              int M, int N, int K, int relu) {
  int lane = threadIdx.x & 31;
  int half = lane >> 4;
  int l16  = lane & 15;
  int mBase = (blockIdx.y * 4 + threadIdx.y) * 16;
  int nBase = blockIdx.x * 16;
  if (mBase >= M) return;

  const float* Arow = A + (size_t)(mBase + l16) * K;
  v8f acc = {0.f, 0.f, 0.f, 0.f, 0.f, 0.f, 0.f, 0.f};

#pragma unroll 4
  for (int k = 0; k < K; k += 4) {
    v2f a, b;
    int ka = k + 2 * half;
    a.x = Arow[ka];
    a.y = Arow[ka + 1];
    const float* Bp = B + (size_t)ka * N + nBase + l16;
    b.x = Bp[0];
    b.y = Bp[N];
    acc = __builtin_amdgcn_wmma_f32_16x16x4_f32(false, a, false, b,
                                                (short)0, acc, false, false);
  }

  float bv = bias ? bias[nBase + l16] : 0.f;
#pragma unroll
  for (int r = 0; r < 8; ++r) {
    float v = acc[r] + bv;
    if (relu) v = fmaxf(v, 0.f);
    D[(size_t)(mBase + r + 8 * half) * N + nBase + l16] = v;
  }
}

// ---------------- rank-1 term: t = P2(L) @ 1 , s = A^T t ----------------
__global__ void tvec_kernel(const float* __restrict__ L1, const float* __restrict__ L2,
                            const float* __restrict__ L3, const float* __restrict__ L4,
                            const float* __restrict__ cf, float* __restrict__ t) {
  int wave = (blockIdx.x * blockDim.x + threadIdx.x) >> 5;
  int lane = threadIdx.x & 31;
  if (wave >= NG) return;
  size_t base = (size_t)wave * NG;
  float p1 = cf[9], p2 = cf[10], p3 = cf[11], p4 = cf[12];
  float acc = 0.f;
  for (int j = lane; j < NG; j += 32) {
    acc = fmaf(p1, L1[base + j], acc);
    acc = fmaf(p2, L2[base + j], acc);
    acc = fmaf(p3, L3[base + j], acc);
    acc = fmaf(p4, L4[base + j], acc);
  }
  acc = waveSum(acc);
  if (lane == 0) t[wave] = cf[8] + acc;
}

__global__ void svec_kernel(const float* __restrict__ A, const float* __restrict__ t,
                            float* __restrict__ s) {
  int j = blockIdx.x * blockDim.x + threadIdx.x;
  if (j >= NG) return;
  float acc = 0.f;
  for (int i = 0; i < NG; ++i) acc = fmaf(A[(size_t)i * NG + j], t[i], acc);
  s[j] = acc;
}

// ---------------- combine G = g0*I + g1*L + ... + g4*L4 (in L4) ----------
__global__ void combineG_kernel(const float* __restrict__ L1, const float* __restrict__ L2,
                                const float* __restrict__ L3, float* __restrict__ L4G,
                                const float* __restrict__ cf) {
  size_t idx = (size_t)blockIdx.x * blockDim.x + threadIdx.x;
  if (idx >= (size_t)NG * NG) return;
  int i = (int)(idx >> 12);
  int j = (int)(idx & (NG - 1));
  float g = cf[1] * L1[idx] + cf[2] * L2[idx] + cf[3] * L3[idx] + cf[4] * L4G[idx];
  if (i == j) g += cf[0];
  L4G[idx] = g;
}

// ---------------- fused (+s_j) row softmax, in place ----------------
__global__ void softmax_rows_kernel(float* __restrict__ C, const float* __restrict__ s) {
  __shared__ float sm[8];
  int row = blockIdx.x;
  int tid = threadIdx.x;
  float* r = C + (size_t)row * NG;

  float mx = -INFINITY;
  for (int j = tid; j < NG; j += blockDim.x) mx = fmaxf(mx, r[j] + s[j]);
  mx = waveMax(mx);
  if ((tid & 31) == 0) sm[tid >> 5] = mx;
  __syncthreads();
  if (tid < 8) { float v = sm[tid]; v = waveMax(v); if (tid == 0) sm[0] = v; }
  __syncthreads();
  mx = sm[0];
  __syncthreads();

  float sum = 0.f;
  for (int j = tid; j < NG; j += blockDim.x) {
    float e = __expf(r[j] + s[j] - mx);
    r[j] = e;
    sum += e;
  }
  sum = waveSum(sum);
  if ((tid & 31) == 0) sm[tid >> 5] = sum;
  __syncthreads();
  if (tid < 8) { float v = sm[tid]; v = waveSum(v); if (tid == 0) sm[0] = v; }
  __syncthreads();
  float inv = 1.0f / sm[0];
  for (int j = tid; j < NG; j += blockDim.x) r[j] *= inv;
}

// ---------------- MaxPool1D(8) along columns ----------------
__global__ void maxpool_kernel(const float* __restrict__ S, float* __restrict__ P) {
  size_t idx = (size_t)blockIdx.x * blockDim.x + threadIdx.x;
  constexpr int NP = NG / POOL;
  if (idx >= (size_t)NG * NP) return;
  int i = (int)(idx / NP);
  int p = (int)(idx % NP);
  const float* r = S + (size_t)i * NG + (size_t)p * POOL;
  float m = r[0];
#pragma unroll
  for (int q = 1; q < POOL; ++q) m = fmaxf(m, r[q]);
  P[idx] = m;
}

// ---------------- final head: [gcn;g2] @ W_f + b_f -> log_softmax ----------
__global__ void final_kernel(const float* __restrict__ gcn, const float* __restrict__ g2,
                             const float* __restrict__ Wf, const float* __restrict__ bf,
                             float* __restrict__ out) {
  int r = blockIdx.x * blockDim.x + threadIdx.x;
  int total = NG + BAT;
  if (r >= total) return;
  const float* x = (r < NG) ? (gcn + (size_t)r * 32) : (g2 + (size_t)(r - NG) * 32);
  float z[NC];
#pragma unroll
  for (int c = 0; c < NC; ++c) {
    float a = bf[c];
#pragma unroll
    for (int k = 0; k < 32; ++k) a = fmaf(x[k], Wf[k * NC + c], a);
    z[c] = a;
  }
  float mx = z[0];
#pragma unroll
  for (int c = 1; c < NC; ++c) mx = fmaxf(mx, z[c]);
  float sum = 0.f;
#pragma unroll
  for (int c = 0; c < NC; ++c) sum += __expf(z[c] - mx);
  float lse = mx + __logf(sum);
#pragma unroll
  for (int c = 0; c < NC; ++c) out[(size_t)r * NC + c] = z[c] - lse;
}

// ===========================================================================
extern "C" void kernel_launch(void* const* d_in, const int* in_sizes, int n_in,
                              void* d_out, int out_size, void* d_ws, size_t ws_size,
                              hipStream_t stream) {
  (void)in_sizes; (void)n_in; (void)out_size; (void)ws_size;

  const float* adj   = (const float*)d_in[0];
  const float* gexp  = (const float*)d_in[1];
  const float* beta  = (const float*)d_in[2];
  const float* W_enc = (const float*)d_in[3];
  const float* b_enc = (const float*)d_in[4];
  const float* W_dec = (const float*)d_in[5];
  const float* b_dec = (const float*)d_in[6];
  const float* W_g1  = (const float*)d_in[7];
  const float* b_g1  = (const float*)d_in[8];
  const float* W_g2  = (const float*)d_in[9];
  const float* b_g2  = (const float*)d_in[10];
  const float* W_f   = (const float*)d_in[11];
  const float* b_f   = (const float*)d_in[12];

  float* out_dec   = (float*)d_out;                    // [4096,4096]
  float* out_final = (float*)d_out + (size_t)NG * NG;  // [4224,10]

  float* ws = (float*)d_ws;
  const size_t NN = (size_t)NG * NG;
  float* L      = ws;
  float* L2     = L  + NN;
  float* L3     = L2 + NN;
  float* L4     = L3 + NN;            // becomes G in place
  float* C      = L4 + NN;            // conv -> softmax in place
  float* pooled = C + NN;             // [N, N/8]
  float* gcn    = pooled + NN / POOL; // [N, 32]
  float* g1m    = gcn + (size_t)NG * 32;
  float* g2m    = g1m + (size_t)BAT * 256;
  float* tvec   = g2m + (size_t)BAT * 32;
  float* svec   = tvec + NG;
  float* dinv   = svec + NG;
  float* dsq    = dinv + NG;
  float* pv     = dsq + NG;
  float* py     = pv + NG;
  float* norm2  = py + NG;            // [NPOW]
  float* cf     = norm2 + 64;         // [16]

  const int T = 256;
  const int ewBlocksNN = (int)((NN + T - 1) / T);
  const int rowWaves   = (NG * 32 + T - 1) / T;

  // 1) degree -> D^{+-1/2}; build L
  deg_kernel<<<rowWaves, T, 0, stream>>>(adj, dinv, dsq);
  buildL_kernel<<<ewBlocksNN, T, 0, stream>>>(adj, dinv, dsq, L);

  // 2) power iteration for lambda_max
  fill_kernel<<<(NG + T - 1) / T, T, 0, stream>>>(pv, 1.0f, NG);
  fill_kernel<<<1, 64, 0, stream>>>(norm2, 0.0f, NPOW);
  for (int it = 0; it < NPOW; ++it) {
    matvec_norm_kernel<<<rowWaves, T, 0, stream>>>(L, pv, py, norm2 + it);
    vec_scale_kernel<<<(NG + T - 1) / T, T, 0, stream>>>(py, pv, norm2 + it);
  }

  // 3) Chebyshev beta -> coefficients in powers of L
  coef_kernel<<<1, 1, 0, stream>>>(beta, norm2 + (NPOW - 1), cf);

  // 4) matrix powers via tiled WMMA GEMM
  {
    dim3 blk(128);
    dim3 grd(NG / TBN, NG / TBM);
    wmma_gemm_tiled<<<grd, blk, 0, stream>>>(L,  L,  L2, nullptr, NG, NG, NG, 0);
    wmma_gemm_tiled<<<grd, blk, 0, stream>>>(L2, L,  L3, nullptr, NG, NG, NG, 0);
    wmma_gemm_tiled<<<grd, blk, 0, stream>>>(L2, L2, L4, nullptr, NG, NG, NG, 0);
  }

  // 5) rank-1 term vectors (before G overwrites L4)
  tvec_kernel<<<rowWaves, T, 0, stream>>>(L, L2, L3, L4, cf, tvec);
  svec_kernel<<<(NG + T - 1) / T, T, 0, stream>>>(adj, tvec, svec);

  // 6) G = P1(L) - P2(L), in place into L4
  combineG_kernel<<<ewBlocksNN, T, 0, stream>>>(L, L2, L3, L4, cf);

  // 7) conv = G @ A
  {
    dim3 blk(128);
    dim3 grd(NG / TBN, NG / TBM);
    wmma_gemm_tiled<<<grd, blk, 0, stream>>>(L4, adj, C, nullptr, NG, NG, NG, 0);
  }

  // 8) softmax over rows of (conv + u s^T), u = ones proxy
  softmax_rows_kernel<<<NG, T, 0, stream>>>(C, svec);

  // 9) maxpool(8)
  maxpool_kernel<<<(int)((NN / POOL + T - 1) / T), T, 0, stream>>>(C, pooled);

  // 10) gcn_out = relu(pooled @ W_enc + b_enc)   [4096,512]x[512,32]
  {
    dim3 blk(32, 4);
    dim3 grd(32 / 16, NG / 64);
    wmma_gemm_f32<<<grd, blk, 0, stream>>>(pooled, W_enc, gcn, b_enc, NG, 32, NG / POOL, 1);
  }

  // 11) decoder_out = relu(gcn @ W_dec + b_dec)  [4096,32]x[32,4096] -> d_out
  {
    dim3 blk(128);
    dim3 grd(NG / TBN, NG / TBM);
    wmma_gemm_tiled<<<grd, blk, 0, stream>>>(gcn, W_dec, out_dec, b_dec, NG, NG, 32, 1);
  }

  // 12) g1 = relu(gene_exp @ W_g1 + b_g1)        [128,4096]x[4096,256]
  {
    dim3 blk(128);
    dim3 grd(256 / TBN, BAT / TBM);
    wmma_gemm_tiled<<<grd, blk, 0, stream>>>(gexp, W_g1, g1m, b_g1, BAT, 256, NG, 1);
  }

  // 13) g2 = relu(g1 @ W_g2 + b_g2)              [128,256]x[256,32]
  {
    dim3 blk(32, 4);
    dim3 grd(32 / 16, BAT / 64);
    wmma_gemm_f32<<<grd, blk, 0, stream>>>(g1m, W_g2, g2m, b_g2, BAT, 32, 256, 1);
  }

  // 14) final log-softmax head -> d_out tail
  final_kernel<<<(NG + BAT + T - 1) / T, T, 0, stream>>>(gcn, g2m, W_f, b_f, out_final);
}